// EncoderTransformer_31791347925256
// MI455X (gfx1250) — compile-verified
//
#include <hip/hip_runtime.h>
#include <hip/hip_bf16.h>

typedef float v2f __attribute__((ext_vector_type(2)));
typedef float v8f __attribute__((ext_vector_type(8)));

#define B_    16
#define S_    1024
#define D_    512
#define NODE_ 32
#define L_    128
#define DK_   64
#define N_    (B_ * NODE_)                 // 512 bags
#define OUT_NODES (N_ * D_)                // 262144 floats, then hidden_flat

// ---------------------------------------------------------------------------
// prep: Wt[n][k] = W_pre[k][n]  (so WMMA B-fragments are contiguous float2),
//       and copy hidden_flat ([2,B,D/2] -> [B,D]) to the output tail.
// ---------------------------------------------------------------------------
__global__ void __launch_bounds__(256)
prep_kernel(const float* __restrict__ W_pre, const float* __restrict__ hidden,
            float* __restrict__ Wt, float* __restrict__ out_hidden)
{
    int e = blockIdx.x * 256 + threadIdx.x;      // 0 .. 32767  (= 512*64)
    int k = e >> 6;                               // 0..511
    int n = e & 63;                               // 0..63
    Wt[n * D_ + k] = W_pre[e];                    // W_pre flat = k*64 + n = e

    if (e < B_ * D_) {                            // 8192 hidden_flat elements
        int b = e >> 9;
        int d = e & (D_ - 1);
        float v = (d < D_ / 2) ? hidden[b * (D_ / 2) + d]
                               : hidden[B_ * (D_ / 2) + b * (D_ / 2) + (d - D_ / 2)];
        out_hidden[e] = v;
    }
}

// ---------------------------------------------------------------------------
// q kernel: qa[b][j] = hidden_flat[b] . W_q[:,j]   (B=16 blocks x 64 threads)
//           epad[b]  = sum_j tanh(b_pre[j]+qa[b][j]) * v_att[j]
// ---------------------------------------------------------------------------
__global__ void __launch_bounds__(64)
q_kernel(const float* __restrict__ hidden, const float* __restrict__ W_q,
         const float* __restrict__ b_pre,  const float* __restrict__ v_att,
         float* __restrict__ qa, float* __restrict__ epad)
{
    const int b = blockIdx.x;
    const int j = threadIdx.x;          // 0..63
    const float* h0 = hidden + b * (D_ / 2);                    // hidden[0][b][:]
    const float* h1 = hidden + B_ * (D_ / 2) + b * (D_ / 2);    // hidden[1][b][:]

    float acc = 0.f;
    for (int d = 0; d < D_ / 2; ++d) acc += h0[d] * W_q[d * DK_ + j];
    for (int d = 0; d < D_ / 2; ++d) acc += h1[d] * W_q[(D_ / 2 + d) * DK_ + j];
    qa[b * DK_ + j] = acc;

    __shared__ float red[64];
    red[j] = tanhf(acc + b_pre[j]) * v_att[j];
    __syncthreads();
    for (int s = 32; s > 0; s >>= 1) {
        if (j < s) red[j] += red[j + s];
        __syncthreads();
    }
    if (j == 0) epad[b] = red[0];
}

// ---------------------------------------------------------------------------
// Main attention kernel: one block per bag (N=512), 128 threads = 4 waves.
// Wave w owns TWO 16-row A-tiles (rows 32w..32w+31) sharing one set of
// B-fragments per k-step: 2 A loads + 4 B loads + 8 WMMA. Halves W-traffic.
// ---------------------------------------------------------------------------
__global__ void __launch_bounds__(128)
attn_kernel(const float* __restrict__ seq,     const int* __restrict__ index,
            const int* __restrict__ lengths,   const float* __restrict__ b_pre,
            const float* __restrict__ v_att,   const float* __restrict__ Wt,
            const float* __restrict__ qa,      const float* __restrict__ epad,
            float* __restrict__ out)
{
    __shared__ int   sIdx[L_];
    __shared__ float sEng[L_];
    __shared__ float sRed[L_];
    __shared__ int   sLen;

    const int n    = blockIdx.x;
    const int b    = n / NODE_;
    const int tid  = threadIdx.x;       // 0..127
    const int wave = tid >> 5;          // 0..3
    const int lane = tid & 31;
    const int half = lane >> 4;         // 0: K+0..1 half, 1: K+2..3 half
    const int l16  = lane & 15;

    sIdx[tid] = index[(size_t)n * L_ + tid];
    if (tid == 0) sLen = lengths[n];
    __syncthreads();

    // ---- WMMA phase: pre[32w..32w+31, 0..63] = bag_tiles @ W_pre ----
    // A layout (16x4 f32): lanes 0-15 hold (M=lane, K=k,k+1), lanes 16-31 (M, K=k+2,k+3)
    const int row0 = (wave << 5) + l16;          // tile 2w
    const int row1 = row0 + 16;                  // tile 2w+1
    const int koff = half ? 2 : 0;
    const float* pA0 = seq + ((size_t)b * S_ + sIdx[row0]) * D_ + koff;
    const float* pA1 = seq + ((size_t)b * S_ + sIdx[row1]) * D_ + koff;
    const float* pB0 = Wt + (size_t)(0 * 16 + l16) * D_ + koff;
    const float* pB1 = Wt + (size_t)(1 * 16 + l16) * D_ + koff;
    const float* pB2 = Wt + (size_t)(2 * 16 + l16) * D_ + koff;
    const float* pB3 = Wt + (size_t)(3 * 16 + l16) * D_ + koff;

    v8f a00 = {}, a01 = {}, a02 = {}, a03 = {};   // rows 32w..32w+15, cols 0..63
    v8f a10 = {}, a11 = {}, a12 = {}, a13 = {};   // rows 32w+16..32w+31
#pragma unroll 2
    for (int k = 0; k < D_; k += 4) {
        v2f fa0 = *(const v2f*)(pA0 + k);
        v2f fa1 = *(const v2f*)(pA1 + k);
        v2f w0  = *(const v2f*)(pB0 + k);
        v2f w1  = *(const v2f*)(pB1 + k);
        v2f w2  = *(const v2f*)(pB2 + k);
        v2f w3  = *(const v2f*)(pB3 + k);
        a00 = __builtin_amdgcn_wmma_f32_16x16x4_f32(false, fa0, false, w0, (short)0, a00, false, false);
        a01 = __builtin_amdgcn_wmma_f32_16x16x4_f32(false, fa0, false, w1, (short)0, a01, false, false);
        a02 = __builtin_amdgcn_wmma_f32_16x16x4_f32(false, fa0, false, w2, (short)0, a02, false, false);
        a03 = __builtin_amdgcn_wmma_f32_16x16x4_f32(false, fa0, false, w3, (short)0, a03, false, false);
        a10 = __builtin_amdgcn_wmma_f32_16x16x4_f32(false, fa1, false, w0, (short)0, a10, false, false);
        a11 = __builtin_amdgcn_wmma_f32_16x16x4_f32(false, fa1, false, w1, (short)0, a11, false, false);
        a12 = __builtin_amdgcn_wmma_f32_16x16x4_f32(false, fa1, false, w2, (short)0, a12, false, false);
        a13 = __builtin_amdgcn_wmma_f32_16x16x4_f32(false, fa1, false, w3, (short)0, a13, false, false);
    }

    // ---- energy: e[row] = sum_col tanh(pre + q + b_pre) * v_att ----
    const int c0 = 0 * 16 + l16, c1 = 1 * 16 + l16, c2 = 2 * 16 + l16, c3 = 3 * 16 + l16;
    const float qb0 = qa[b * DK_ + c0] + b_pre[c0], va0 = v_att[c0];
    const float qb1 = qa[b * DK_ + c1] + b_pre[c1], va1 = v_att[c1];
    const float qb2 = qa[b * DK_ + c2] + b_pre[c2], va2 = v_att[c2];
    const float qb3 = qa[b * DK_ + c3] + b_pre[c3], va3 = v_att[c3];

    float g0[8], g1[8];
#pragma unroll
    for (int j = 0; j < 8; ++j) {
        g0[j] = tanhf(a00[j] + qb0) * va0 + tanhf(a01[j] + qb1) * va1 +
                tanhf(a02[j] + qb2) * va2 + tanhf(a03[j] + qb3) * va3;
        g1[j] = tanhf(a10[j] + qb0) * va0 + tanhf(a11[j] + qb1) * va1 +
                tanhf(a12[j] + qb2) * va2 + tanhf(a13[j] + qb3) * va3;
    }
    // reduce across each 16-lane half (C layout: lanes 0-15 -> row j, 16-31 -> row j+8)
#pragma unroll
    for (int j = 0; j < 8; ++j) {
#pragma unroll
        for (int o = 1; o < 16; o <<= 1) {
            g0[j] += __shfl_xor(g0[j], o);
            g1[j] += __shfl_xor(g1[j], o);
        }
    }
    if (l16 == 0) {
        const int base = (wave << 5) + (half << 3);
#pragma unroll
        for (int j = 0; j < 8; ++j) {
            sEng[base + j]      = g0[j];
            sEng[base + 16 + j] = g1[j];
        }
    }
    __syncthreads();

    // ---- softmax over L (rows >= len contribute the constant e_pad) ----
    const float ep = epad[b];
    const int  len = sLen;
    {
        float e = (tid < len) ? sEng[tid] : ep;
        sEng[tid] = e;
        sRed[tid] = e;
    }
    __syncthreads();
    for (int s = L_ / 2; s > 0; s >>= 1) {
        if (tid < s) sRed[tid] = fmaxf(sRed[tid], sRed[tid + s]);
        __syncthreads();
    }
    const float m = sRed[0];
    __syncthreads();
    {
        float ex = __expf(sEng[tid] - m);
        sEng[tid] = ex;            // unnormalized score
        sRed[tid] = ex;
    }
    __syncthreads();
    for (int s = L_ / 2; s > 0; s >>= 1) {
        if (tid < s) sRed[tid] += sRed[tid + s];
        __syncthreads();
    }
    const float inv = 1.0f / sRed[0];

    // ---- context: out[n,:] = (sum_{l<len} score_l * bag_l) / sum ----
    float ctx0 = 0.f, ctx1 = 0.f, ctx2 = 0.f, ctx3 = 0.f;
    for (int l = 0; l < len; ++l) {
        const float  sc   = sEng[l];
        const float* rowp = seq + ((size_t)b * S_ + sIdx[l]) * D_;
        int lp = (l + 2 < len) ? (l + 2) : l;                // uniform per block
        const float* pf = seq + ((size_t)b * S_ + sIdx[lp]) * D_;
        __builtin_prefetch(pf + tid, 0, 3);
        ctx0 += sc * rowp[tid];
        ctx1 += sc * rowp[tid + 128];
        ctx2 += sc * rowp[tid + 256];
        ctx3 += sc * rowp[tid + 384];
    }
    float* op = out + (size_t)n * D_;
    op[tid]       = ctx0 * inv;
    op[tid + 128] = ctx1 * inv;
    op[tid + 256] = ctx2 * inv;
    op[tid + 384] = ctx3 * inv;
}

// ---------------------------------------------------------------------------
extern "C" void kernel_launch(void* const* d_in, const int* in_sizes, int n_in,
                              void* d_out, int out_size, void* d_ws, size_t ws_size,
                              hipStream_t stream)
{
    const float* seq     = (const float*)d_in[0];
    const float* hidden  = (const float*)d_in[1];
    const int*   index   = (const int*)  d_in[2];
    const int*   lengths = (const int*)  d_in[3];
    const float* W_pre   = (const float*)d_in[4];
    const float* b_pre   = (const float*)d_in[5];
    const float* W_q     = (const float*)d_in[6];
    const float* v_att   = (const float*)d_in[7];
    float* out = (float*)d_out;

    float* ws   = (float*)d_ws;
    float* Wt   = ws;                      // 64*512  = 32768 floats
    float* qa   = ws + 32768;              // 16*64   =  1024 floats
    float* epad = ws + 32768 + 1024;       // 16 floats

    prep_kernel<<<(D_ * DK_) / 256, 256, 0, stream>>>(W_pre, hidden, Wt, out + OUT_NODES);
    q_kernel<<<B_, DK_, 0, stream>>>(hidden, W_q, b_pre, v_att, qa, epad);
    attn_kernel<<<N_, 128, 0, stream>>>(seq, index, lengths, b_pre, v_att, Wt, qa, epad, out);
}